// Router_18468359373121
// MI455X (gfx1250) — compile-verified
//
#include <hip/hip_runtime.h>
#include <cstdint>
#include <cstddef>

// ---------------------------------------------------------------------------
// MoE router for MI455X (gfx1250, wave32).
//   logits = h @ W^T   (f32, via V_WMMA_F32_16X16X4_F32)
//   mask   = top8(logits + b63*e63)  -> out[0 .. TE)     (0.0 / 1.0)
//   probs  = softmax(logits)         -> out[TE .. 2TE)
//   logits                           -> out[2TE .. 3TE)
// Double-buffered GLOBAL_LOAD_ASYNC_TO_LDS staging when the toolchain
// exposes the gfx1250 async builtins; synchronous staging otherwise.
// ---------------------------------------------------------------------------

typedef __attribute__((ext_vector_type(2))) float v2f;
typedef __attribute__((ext_vector_type(8))) float v8f;
typedef int v4i __attribute__((vector_size(4 * sizeof(int))));

#define TOPK 8

#define BLOCK 256
#define WAVES (BLOCK / 32)
#define ROWS_PER_WAVE 16
#define ROWS_PER_WG (WAVES * ROWS_PER_WAVE)   // 128
#define KC 64                                  // K-chunk staged in LDS
#define LDA 68                                 // padded LDS row stride (floats)
#define LDL 65                                 // padded logits stride (floats)
#define BUF_FLOATS ((ROWS_PER_WG + 64) * LDA)  // one chunk buffer (h + W tiles)
#define ASYNC_PER_THREAD 12                    // 8 (h tile) + 4 (W tile) B128 ops

#if defined(__HIP_DEVICE_COMPILE__) && __has_builtin(__builtin_amdgcn_global_load_async_to_lds_b128)
#define HAVE_ASYNC_LDS 1
#else
#define HAVE_ASYNC_LDS 0
#endif

#if HAVE_ASYNC_LDS
__device__ __forceinline__ void async_load_b128(const float* g, float* l) {
  __builtin_amdgcn_global_load_async_to_lds_b128(
      (__attribute__((address_space(1))) v4i*)(g),
      (__attribute__((address_space(3))) v4i*)(l),
      /*offset=*/0, /*cpol=*/0);
}
#endif

template <int N>
__device__ __forceinline__ void wait_asynccnt() {
#if defined(__HIP_DEVICE_COMPILE__) && __has_builtin(__builtin_amdgcn_s_wait_asynccnt)
  __builtin_amdgcn_s_wait_asynccnt((unsigned short)N);
#elif defined(__HIP_DEVICE_COMPILE__)
  asm volatile("s_wait_asynccnt %0" :: "i"(N) : "memory");
#endif
}

// Issue one chunk's staging (async path).
__device__ __forceinline__ void stage_chunk_async(
    const float* __restrict__ h, const float* __restrict__ W,
    float* bufA, float* bufB, int row0, int kc, int D, int tid) {
#if HAVE_ASYNC_LDS
#pragma unroll
  for (int j = 0; j < 8; ++j) {                 // h tile: 128 x 64 f32
    const int i  = tid + j * BLOCK;
    const int r  = i >> 4;
    const int c4 = (i & 15) << 2;
    async_load_b128(h + (size_t)(row0 + r) * D + kc + c4, bufA + r * LDA + c4);
  }
#pragma unroll
  for (int j = 0; j < 4; ++j) {                 // W tile: 64 x 64 f32
    const int i  = tid + j * BLOCK;
    const int r  = i >> 4;
    const int c4 = (i & 15) << 2;
    async_load_b128(W + (size_t)r * D + kc + c4, bufB + r * LDA + c4);
  }
#endif
}

// Synchronous staging fallback.
__device__ __forceinline__ void stage_chunk_sync(
    const float* __restrict__ h, const float* __restrict__ W,
    float* bufA, float* bufB, int row0, int kc, int D, int tid) {
#pragma unroll
  for (int j = 0; j < 8; ++j) {
    const int i  = tid + j * BLOCK;
    const int r  = i >> 4;
    const int c4 = (i & 15) << 2;
    *(float4*)(bufA + r * LDA + c4) =
        *(const float4*)(h + (size_t)(row0 + r) * D + kc + c4);
    if (kc + KC < D)
      __builtin_prefetch(h + (size_t)(row0 + r) * D + (kc + KC) + c4, 0, 0);
  }
#pragma unroll
  for (int j = 0; j < 4; ++j) {
    const int i  = tid + j * BLOCK;
    const int r  = i >> 4;
    const int c4 = (i & 15) << 2;
    *(float4*)(bufB + r * LDA + c4) =
        *(const float4*)(W + (size_t)r * D + kc + c4);
  }
}

__global__ __launch_bounds__(BLOCK) void router_wmma_kernel(
    const float* __restrict__ h,     // [T, D]
    const float* __restrict__ W,     // [E, D]
    const float* __restrict__ bias,  // [E]
    float* __restrict__ out,         // [3, T, E] flat
    int T, int D, int E)
{
  extern __shared__ float smem[];
  float* A0 = smem;                        // buffer 0: h tile
  float* B0 = smem + ROWS_PER_WG * LDA;    // buffer 0: W tile
  float* A1 = smem + BUF_FLOATS;           // buffer 1: h tile
  float* B1 = A1 + ROWS_PER_WG * LDA;      // buffer 1: W tile
  float* sL = smem;                        // logits (aliases buffer 0 later)

  const int tid  = threadIdx.x;
  const int lane = tid & 31;               // wave32
  const int wave = tid >> 5;
  const int half = lane >> 4;              // 0: lanes 0-15, 1: lanes 16-31
  const int lm   = lane & 15;
  const int row0 = blockIdx.x * ROWS_PER_WG;
  const int wrow = wave * ROWS_PER_WAVE;

  v8f acc0 = {}, acc1 = {}, acc2 = {}, acc3 = {};

  const int nch = D / KC;                  // 32 chunks

#if HAVE_ASYNC_LDS
  stage_chunk_async(h, W, A0, B0, row0, 0, D, tid);
#endif

  for (int c = 0; c < nch; ++c) {
    float* cA = (c & 1) ? A1 : A0;
    float* cB = (c & 1) ? B1 : B0;
#if HAVE_ASYNC_LDS
    // Prefetch next chunk into the other buffer (its readers finished at the
    // barrier that ended iteration c-1), then wait for the current buffer's
    // loads (in-order completion) and make them workgroup-visible.
    if (c + 1 < nch) {
      stage_chunk_async(h, W, (c & 1) ? A0 : A1, (c & 1) ? B0 : B1,
                        row0, (c + 1) * KC, D, tid);
      wait_asynccnt<ASYNC_PER_THREAD>();
    } else {
      wait_asynccnt<0>();
    }
    __syncthreads();
#else
    __syncthreads();
    stage_chunk_sync(h, W, cA, cB, row0, c * KC, D, tid);
    __syncthreads();
#endif

    // -------- WMMA over the chunk: 16 k-quads ----------------------------
    // A 16x4 f32 layout (ISA 7.12.2): lanes 0-15 hold K0(v0),K1(v1);
    // lanes 16-31 hold K2(v0),K3(v1). B symmetric (4x16, N in lanes).
#pragma unroll
    for (int k4 = 0; k4 < KC / 4; ++k4) {
      const int k = (k4 << 2) + (half << 1);
      const v2f a  = *(const v2f*)(cA + (wrow + lm) * LDA + k);
      const v2f b0 = *(const v2f*)(cB + ( 0 + lm) * LDA + k);
      const v2f b1 = *(const v2f*)(cB + (16 + lm) * LDA + k);
      const v2f b2 = *(const v2f*)(cB + (32 + lm) * LDA + k);
      const v2f b3 = *(const v2f*)(cB + (48 + lm) * LDA + k);
      acc0 = __builtin_amdgcn_wmma_f32_16x16x4_f32(false, a, false, b0, (short)0, acc0, false, false);
      acc1 = __builtin_amdgcn_wmma_f32_16x16x4_f32(false, a, false, b1, (short)0, acc1, false, false);
      acc2 = __builtin_amdgcn_wmma_f32_16x16x4_f32(false, a, false, b2, (short)0, acc2, false, false);
      acc3 = __builtin_amdgcn_wmma_f32_16x16x4_f32(false, a, false, b3, (short)0, acc3, false, false);
    }
    __syncthreads();   // all waves done reading before buffer is overwritten
  }

  // -------- spill logits to LDS (C layout: VGPR r -> M=r or r+8) ----------
#pragma unroll
  for (int r = 0; r < 8; ++r) {
    const int M = r + (half << 3);
    float* dst = sL + (wrow + M) * LDL + lm;
    dst[ 0] = acc0[r];
    dst[16] = acc1[r];
    dst[32] = acc2[r];
    dst[48] = acc3[r];
  }
  __syncthreads();

  // -------- per-row epilogue: softmax + top-8 + outputs --------------------
  if (tid < ROWS_PER_WG) {
    const float* rl   = sL + tid * LDL;
    const size_t grow = (size_t)row0 + tid;
    const size_t TE   = (size_t)T * 64;
    float* maskO = out;
    float* probO = out + TE;
    float* logO  = out + 2 * TE;

    float mx = rl[0];
#pragma unroll 1
    for (int e = 1; e < 64; ++e) mx = fmaxf(mx, rl[e]);
    float sum = 0.0f;
#pragma unroll 1
    for (int e = 0; e < 64; ++e) sum += __expf(rl[e] - mx);
    const float inv = 1.0f / sum;

#pragma unroll 1
    for (int e4 = 0; e4 < 64; e4 += 4) {
      const float l0 = rl[e4 + 0], l1 = rl[e4 + 1], l2 = rl[e4 + 2], l3 = rl[e4 + 3];
      *(float4*)(logO  + grow * 64 + e4) = make_float4(l0, l1, l2, l3);
      *(float4*)(probO + grow * 64 + e4) = make_float4(
          __expf(l0 - mx) * inv, __expf(l1 - mx) * inv,
          __expf(l2 - mx) * inv, __expf(l3 - mx) * inv);
    }

    // top-8 on logits + bias[63] on the last (identity) expert.
    // Strict '>' scanning from e=0 gives lowest-index ties like jax top_k.
    const float b63 = bias[63];
    unsigned long long sel = 0ull;
#pragma unroll 1
    for (int it = 0; it < TOPK; ++it) {
      float best = -3.402823466e+38f;
      int bi = 0;
#pragma unroll 1
      for (int e = 0; e < 64; ++e) {
        if ((sel >> e) & 1ull) continue;
        const float v = rl[e] + (e == 63 ? b63 : 0.0f);
        if (v > best) { best = v; bi = e; }
      }
      sel |= (1ull << bi);
    }
#pragma unroll 1
    for (int e4 = 0; e4 < 64; e4 += 4) {
      *(float4*)(maskO + grow * 64 + e4) = make_float4(
          ((sel >> (e4 + 0)) & 1ull) ? 1.0f : 0.0f,
          ((sel >> (e4 + 1)) & 1ull) ? 1.0f : 0.0f,
          ((sel >> (e4 + 2)) & 1ull) ? 1.0f : 0.0f,
          ((sel >> (e4 + 3)) & 1ull) ? 1.0f : 0.0f);
    }
  }
}

extern "C" void kernel_launch(void* const* d_in, const int* in_sizes, int n_in,
                              void* d_out, int out_size, void* d_ws, size_t ws_size,
                              hipStream_t stream) {
  const float* h    = (const float*)d_in[0];
  const float* W    = (const float*)d_in[1];
  const float* bias = (const float*)d_in[2];
  float* out = (float*)d_out;

  const int E = in_sizes[2];          // 64
  const int D = in_sizes[1] / E;      // 2048
  const int T = in_sizes[0] / D;      // 16384

  const int grid = T / ROWS_PER_WG;   // 128 workgroups of 8 wave32s
  // Always allocate the double-buffer footprint so host/device agree
  // regardless of which staging path the device compiler selected.
  const size_t shmem = (size_t)(2 * BUF_FLOATS) * sizeof(float); // 104448 B

  router_wmma_kernel<<<grid, BLOCK, shmem, stream>>>(h, W, bias, out, T, D, E);
}